// BinaryTreeComposer_66597762891840
// MI455X (gfx1250) — compile-verified
//
#include <hip/hip_runtime.h>
#include <math.h>

#define MEMN   2048
#define KSPLIT 8                 // waves per block; each wave covers MEMN/KSPLIT of K
#define KCHUNK (MEMN / KSPLIT)   // 256

typedef __attribute__((ext_vector_type(2))) float v2f;
typedef __attribute__((ext_vector_type(8))) float v8f;

// D = A(16x4 f32) * B(4x16 f32) + C, full fp32 precision.
__device__ __forceinline__ v8f wmma4(v2f a, v2f b, v8f c) {
  return __builtin_amdgcn_wmma_f32_16x16x4_f32(
      /*neg_a=*/false, a, /*neg_b=*/false, b,
      /*c_mod=*/(short)0, c, /*reuse_a=*/false, /*reuse_b=*/false);
}

__device__ __forceinline__ float sigmf(float x) { return 1.0f / (1.0f + expf(-x)); }

// Coalesced b128 copy of a [MEMN] f32 vector into LDS (done by whole block).
__device__ __forceinline__ void stage(float* dst, const float* __restrict__ src) {
  for (int i = threadIdx.x; i < MEMN / 4; i += 256)
    ((float4*)dst)[i] = ((const float4*)src)[i];
}

// Broadcast-N accumulator: row j0+r lives in acc[r] (lanes 0-15), row j0+8+r in
// acc[r] (lanes 16-31). Stash the 16 row sums into lds[wave][0..15].
__device__ __forceinline__ void stash(float (*red)[16], int wave, int lane, v8f acc) {
  if (lane == 0) {
    #pragma unroll
    for (int r = 0; r < 8; ++r) red[wave][r] = acc[r];
  } else if (lane == 16) {
    #pragma unroll
    for (int r = 0; r < 8; ++r) red[wave][8 + r] = acc[r];
  }
}

__device__ __forceinline__ float sumred(const float (*red)[16], int t) {
  float s = 0.f;
  #pragma unroll
  for (int w = 0; w < KSPLIT; ++w) s += red[w][t];
  return s;
}

// ---------------------------------------------------------------------------
// Kernel 1: m_left = tanh(Wh.lh + Us.S + bWh + bUs); m_right likewise with rh.
// Wh tile is loaded ONCE and feeds two WMMA accumulators (halves Wh traffic).
// B operands come from LDS so the weight stream waits only on LOADcnt.
// ---------------------------------------------------------------------------
__global__ __launch_bounds__(256) void score_gemv(
    const float* __restrict__ Wh, const float* __restrict__ Whb,
    const float* __restrict__ Us, const float* __restrict__ Usb,
    const float* __restrict__ lh, const float* __restrict__ rh,
    const float* __restrict__ S,
    float* __restrict__ mL, float* __restrict__ mR)
{
  __shared__ float xls[MEMN], xrs[MEMN], xss[MEMN];
  __shared__ float rL[KSPLIT][16], rR[KSPLIT][16], rS[KSPLIT][16];
  stage(xls, lh);
  stage(xrs, rh);
  stage(xss, S);
  __syncthreads();

  const int lane = threadIdx.x & 31;
  const int wave = threadIdx.x >> 5;
  const int j0   = blockIdx.x * 16;
  const int mrow = lane & 15;           // M index within the 16-row tile
  const int koff = (lane >> 4) << 1;    // K sub-offset: 0 (lanes 0-15) or 2
  const int kb   = wave * KCHUNK;       // this wave's K slice

  const float* wrow = Wh + (size_t)(j0 + mrow) * MEMN + kb + koff;
  const float* urow = Us + (size_t)(j0 + mrow) * MEMN + kb + koff;
  const float* bl = xls + kb + koff;
  const float* br = xrs + kb + koff;
  const float* bs = xss + kb + koff;

  v8f aL = {}, aR = {}, aS = {};
  #pragma unroll 8
  for (int k = 0; k < KCHUNK; k += 4) {
    v2f aW = *(const v2f*)(wrow + k);     // weight tile (A operand, HBM stream)
    v2f aU = *(const v2f*)(urow + k);
    v2f bL = *(const v2f*)(bl + k);       // broadcast vector (B operand, LDS)
    v2f bR = *(const v2f*)(br + k);
    v2f bS = *(const v2f*)(bs + k);
    aL = wmma4(aW, bL, aL);
    aR = wmma4(aW, bR, aR);
    aS = wmma4(aU, bS, aS);
  }

  stash(rL, wave, lane, aL);
  stash(rR, wave, lane, aR);
  stash(rS, wave, lane, aS);
  __syncthreads();
  if (threadIdx.x < 16) {
    const int t = threadIdx.x, row = j0 + t;
    float s = sumred(rS, t) + Whb[row] + Usb[row];
    mL[row] = tanhf(sumred(rL, t) + s);
    mR[row] = tanhf(sumred(rR, t) + s);
  }
}

// ---------------------------------------------------------------------------
// Kernel 2: eL = w.mL, eR = w.mR ; alphas = {eL, eR} / (eL + eR)
// ---------------------------------------------------------------------------
__global__ __launch_bounds__(256) void attn_reduce(
    const float* __restrict__ w, const float* __restrict__ mL,
    const float* __restrict__ mR, float* __restrict__ alphas)
{
  __shared__ float sL[256], sR[256];
  float aL = 0.f, aR = 0.f;
  for (int i = threadIdx.x; i < MEMN; i += 256) {
    float wv = w[i];
    aL += wv * mL[i];
    aR += wv * mR[i];
  }
  sL[threadIdx.x] = aL; sR[threadIdx.x] = aR;
  __syncthreads();
  for (int s = 128; s > 0; s >>= 1) {
    if (threadIdx.x < s) {
      sL[threadIdx.x] += sL[threadIdx.x + s];
      sR[threadIdx.x] += sR[threadIdx.x + s];
    }
    __syncthreads();
  }
  if (threadIdx.x == 0) {
    float eL = sL[0], eR = sR[0], denom = eL + eR;
    alphas[0] = eL / denom;
    alphas[1] = eR / denom;
  }
}

// ---------------------------------------------------------------------------
// Kernel 3: la = tanh(aL*(ma.lh) + b), ra = tanh(aR*(ma.rh) + b).
// (scalar alpha hoisted out of the GEMV by linearity; ma loaded once)
// ---------------------------------------------------------------------------
__global__ __launch_bounds__(256) void ma_gemv(
    const float* __restrict__ Ma, const float* __restrict__ Mab,
    const float* __restrict__ lh, const float* __restrict__ rh,
    const float* __restrict__ alphas,
    float* __restrict__ la, float* __restrict__ ra)
{
  __shared__ float xls[MEMN], xrs[MEMN];
  __shared__ float rL[KSPLIT][16], rR[KSPLIT][16];
  stage(xls, lh);
  stage(xrs, rh);
  __syncthreads();

  const int lane = threadIdx.x & 31;
  const int wave = threadIdx.x >> 5;
  const int j0   = blockIdx.x * 16;
  const int mrow = lane & 15;
  const int koff = (lane >> 4) << 1;
  const int kb   = wave * KCHUNK;

  const float* wrow = Ma + (size_t)(j0 + mrow) * MEMN + kb + koff;
  const float* bl = xls + kb + koff;
  const float* br = xrs + kb + koff;

  v8f aL = {}, aR = {};
  #pragma unroll 8
  for (int k = 0; k < KCHUNK; k += 4) {
    v2f aW = *(const v2f*)(wrow + k);
    v2f bL = *(const v2f*)(bl + k);
    v2f bR = *(const v2f*)(br + k);
    aL = wmma4(aW, bL, aL);
    aR = wmma4(aW, bR, aR);
  }

  stash(rL, wave, lane, aL);
  stash(rR, wave, lane, aR);
  __syncthreads();
  if (threadIdx.x < 16) {
    const int t = threadIdx.x, row = j0 + t;
    float alphL = alphas[0], alphR = alphas[1], b = Mab[row];
    la[row] = tanhf(alphL * sumred(rL, t) + b);
    ra[row] = tanhf(alphR * sumred(rR, t) + b);
  }
}

// ---------------------------------------------------------------------------
// Kernel 4: 8 gate GEMVs fused with the LSTM cell update.
// out[0:2048] = c, out[2048:4096] = h = tanh(c)
// ---------------------------------------------------------------------------
__global__ __launch_bounds__(256) void gates_gemv(
    const float* __restrict__ Wil,  const float* __restrict__ bil,
    const float* __restrict__ Wir,  const float* __restrict__ bir,
    const float* __restrict__ Wlfl, const float* __restrict__ blfl,
    const float* __restrict__ Wlfr, const float* __restrict__ blfr,
    const float* __restrict__ Wrfl, const float* __restrict__ brfl,
    const float* __restrict__ Wrfr, const float* __restrict__ brfr,
    const float* __restrict__ Wul,  const float* __restrict__ bul,
    const float* __restrict__ Wur,  const float* __restrict__ bur,
    const float* __restrict__ la,   const float* __restrict__ ra,
    const float* __restrict__ lc,   const float* __restrict__ rc,
    float* __restrict__ out)
{
  __shared__ float xls[MEMN], xrs[MEMN];
  __shared__ float red[8][KSPLIT][16];
  stage(xls, la);
  stage(xrs, ra);
  __syncthreads();

  const int lane = threadIdx.x & 31;
  const int wave = threadIdx.x >> 5;
  const int j0   = blockIdx.x * 16;
  const int mrow = lane & 15;
  const int koff = (lane >> 4) << 1;
  const int kb   = wave * KCHUNK;
  const size_t roff = (size_t)(j0 + mrow) * MEMN + kb + koff;

  const float* p0 = Wil  + roff; const float* p1 = Wir  + roff;
  const float* p2 = Wlfl + roff; const float* p3 = Wlfr + roff;
  const float* p4 = Wrfl + roff; const float* p5 = Wrfr + roff;
  const float* p6 = Wul  + roff; const float* p7 = Wur  + roff;
  const float* bl = xls + kb + koff;
  const float* br = xrs + kb + koff;

  v8f a0 = {}, a1 = {}, a2 = {}, a3 = {}, a4 = {}, a5 = {}, a6 = {}, a7 = {};
  #pragma unroll 4
  for (int k = 0; k < KCHUNK; k += 4) {
    v2f bA = *(const v2f*)(bl + k);
    v2f bB = *(const v2f*)(br + k);
    a0 = wmma4(*(const v2f*)(p0 + k), bA, a0);
    a1 = wmma4(*(const v2f*)(p1 + k), bB, a1);
    a2 = wmma4(*(const v2f*)(p2 + k), bA, a2);
    a3 = wmma4(*(const v2f*)(p3 + k), bB, a3);
    a4 = wmma4(*(const v2f*)(p4 + k), bA, a4);
    a5 = wmma4(*(const v2f*)(p5 + k), bB, a5);
    a6 = wmma4(*(const v2f*)(p6 + k), bA, a6);
    a7 = wmma4(*(const v2f*)(p7 + k), bB, a7);
  }

  stash(red[0], wave, lane, a0);
  stash(red[1], wave, lane, a1);
  stash(red[2], wave, lane, a2);
  stash(red[3], wave, lane, a3);
  stash(red[4], wave, lane, a4);
  stash(red[5], wave, lane, a5);
  stash(red[6], wave, lane, a6);
  stash(red[7], wave, lane, a7);
  __syncthreads();
  if (threadIdx.x < 16) {
    const int t = threadIdx.x, row = j0 + t;
    float iv = sigmf(sumred(red[0], t) + sumred(red[1], t) + bil[row]  + bir[row]);
    float lf = sigmf(sumred(red[2], t) + sumred(red[3], t) + blfl[row] + blfr[row]);
    float rf = sigmf(sumred(red[4], t) + sumred(red[5], t) + brfl[row] + brfr[row]);
    float uv = tanhf(sumred(red[6], t) + sumred(red[7], t) + bul[row]  + bur[row]);
    float cv = iv * uv + lf * lc[row] + rf * rc[row];
    out[row]        = cv;
    out[MEMN + row] = tanhf(cv);
  }
}

extern "C" void kernel_launch(void* const* d_in, const int* in_sizes, int n_in,
                              void* d_out, int out_size, void* d_ws, size_t ws_size,
                              hipStream_t stream) {
  (void)in_sizes; (void)n_in; (void)out_size; (void)ws_size;
  const float* lc = (const float*)d_in[0];
  const float* lh = (const float*)d_in[1];
  const float* rc = (const float*)d_in[2];
  const float* rh = (const float*)d_in[3];
  const float* S  = (const float*)d_in[4];
  const float* w  = (const float*)d_in[5];
  const float* Wh_w   = (const float*)d_in[6];  const float* Wh_b   = (const float*)d_in[7];
  const float* Us_w   = (const float*)d_in[8];  const float* Us_b   = (const float*)d_in[9];
  const float* ma_w   = (const float*)d_in[10]; const float* ma_b   = (const float*)d_in[11];
  const float* ilh_w  = (const float*)d_in[12]; const float* ilh_b  = (const float*)d_in[13];
  const float* irh_w  = (const float*)d_in[14]; const float* irh_b  = (const float*)d_in[15];
  const float* lflh_w = (const float*)d_in[16]; const float* lflh_b = (const float*)d_in[17];
  const float* lfrh_w = (const float*)d_in[18]; const float* lfrh_b = (const float*)d_in[19];
  const float* rflh_w = (const float*)d_in[20]; const float* rflh_b = (const float*)d_in[21];
  const float* rfrh_w = (const float*)d_in[22]; const float* rfrh_b = (const float*)d_in[23];
  const float* ulh_w  = (const float*)d_in[24]; const float* ulh_b  = (const float*)d_in[25];
  const float* urh_w  = (const float*)d_in[26]; const float* urh_b  = (const float*)d_in[27];

  float* out = (float*)d_out;
  float* ws  = (float*)d_ws;
  float* mLb    = ws;
  float* mRb    = ws + 1 * MEMN;
  float* laBuf  = ws + 2 * MEMN;
  float* raBuf  = ws + 3 * MEMN;
  float* alphas = ws + 4 * MEMN;   // 2 floats

  const int blocks = MEMN / 16;    // 128 row-tile blocks, 8-way K split each

  score_gemv<<<blocks, 256, 0, stream>>>(Wh_w, Wh_b, Us_w, Us_b, lh, rh, S, mLb, mRb);
  attn_reduce<<<1, 256, 0, stream>>>(w, mLb, mRb, alphas);
  ma_gemv<<<blocks, 256, 0, stream>>>(ma_w, ma_b, lh, rh, alphas, laBuf, raBuf);
  gates_gemv<<<blocks, 256, 0, stream>>>(
      ilh_w, ilh_b, irh_w, irh_b,
      lflh_w, lflh_b, lfrh_w, lfrh_b,
      rflh_w, rflh_b, rfrh_w, rfrh_b,
      ulh_w, ulh_b, urh_w, urh_b,
      laBuf, raBuf, lc, rc, out);
}